// LogReg_30193620091430
// MI455X (gfx1250) — compile-verified
//
#include <hip/hip_runtime.h>
#include <hip/hip_bf16.h>

// Problem constants from the reference (fixed shapes).
#define NUM_GRAPHS      10000
#define NODES_PER_GRAPH 100
#define FT_IN           256
#define NB_CLASSES      128

typedef __attribute__((ext_vector_type(2))) float v2f;
typedef __attribute__((ext_vector_type(8))) float v8f;

// ---------------------------------------------------------------------------
// Kernel 1: per-graph sum pooling.
//   seq is [NUM_GRAPHS*NODES_PER_GRAPH, FT_IN] row-major, each graph's block
//   of 100x256 floats is contiguous (25600 floats = 6400 float4).
//   64 lanes x float4 = one full 1KB feature row per iteration -> perfectly
//   coalesced 16B/lane global_load_b128. 4 graphs per 256-thread block.
// ---------------------------------------------------------------------------
__global__ void __launch_bounds__(256)
pool_sum_kernel(const float* __restrict__ seq, float* __restrict__ pooled) {
    const int g = blockIdx.x * 4 + (threadIdx.x >> 6);   // graph id
    const int t = threadIdx.x & 63;                       // float4 column 0..63

    const float4* __restrict__ p =
        reinterpret_cast<const float4*>(seq) + (size_t)g * (FT_IN / 4) * NODES_PER_GRAPH + t;

    float4 acc = make_float4(0.f, 0.f, 0.f, 0.f);
    #pragma unroll 4
    for (int n = 0; n < NODES_PER_GRAPH; ++n) {
        float4 v = p[n * (FT_IN / 4)];
        acc.x += v.x; acc.y += v.y; acc.z += v.z; acc.w += v.w;
    }
    reinterpret_cast<float4*>(pooled)[(size_t)g * (FT_IN / 4) + t] = acc;
}

// ---------------------------------------------------------------------------
// Kernel 2: out = PReLU(pooled @ W^T + b) via V_WMMA_F32_16X16X4_F32.
//   pooled: [NUM_GRAPHS, FT_IN]   (fp32, in d_ws, hot in L2)
//   W:      [NB_CLASSES, FT_IN]   (fp32, row-major) -> B[k][n] = W[n][k]
//   One 16x16 output tile per wave; block = 8 waves = 8 N-tiles of one M-tile.
//   Grid = NUM_GRAPHS/16 = 625 blocks.
//
//   A-layout (32-bit A 16x4, MxK): lanes 0-15 -> M=lane, K={k0,k0+1};
//                                  lanes 16-31 -> M=lane-16, K={k0+2,k0+3}.
//   B-layout mirrors it with N in place of M.
//   C/D layout: VGPR r -> row r (lanes 0-15) / row r+8 (lanes 16-31),
//               col = lane & 15.
// ---------------------------------------------------------------------------
__global__ void __launch_bounds__(256)
fc_prelu_wmma_kernel(const float* __restrict__ pooled,
                     const float* __restrict__ W,
                     const float* __restrict__ bias,
                     const float* __restrict__ prelu_a,
                     float* __restrict__ out) {
    const int wave  = threadIdx.x >> 5;   // 0..7 -> N tile
    const int lane  = threadIdx.x & 31;
    const int mtile = blockIdx.x;         // 0..624
    const int row16 = lane & 15;          // M index for A, N index for B
    const int khalf = (lane >> 4) << 1;   // 0 for lanes 0-15, 2 for lanes 16-31

    const float* __restrict__ Arow = pooled + ((size_t)mtile * 16 + row16) * FT_IN + khalf;
    const float* __restrict__ Brow = W      + ((size_t)wave  * 16 + row16) * FT_IN + khalf;

    v8f acc = {};
    #pragma unroll 8
    for (int k0 = 0; k0 < FT_IN; k0 += 4) {
        // 8-byte aligned contiguous pair loads (global_load_b64 per operand)
        v2f a = *reinterpret_cast<const v2f*>(Arow + k0);
        v2f b = *reinterpret_cast<const v2f*>(Brow + k0);
        // 8 args: (neg_a, A, neg_b, B, c_mod, C, reuse_a, reuse_b)
        acc = __builtin_amdgcn_wmma_f32_16x16x4_f32(
            false, a, false, b, (short)0, acc, false, false);
    }

    // Epilogue: bias + scalar-slope PReLU, then store.
    const float slope = prelu_a[0];
    const int   ncol  = lane & 15;
    const float bval  = bias[wave * 16 + ncol];
    const int   mbase = mtile * 16 + ((lane >> 4) << 3);  // +8 rows for upper half-wave

    #pragma unroll
    for (int r = 0; r < 8; ++r) {
        float v = acc[r] + bval;
        v = (v >= 0.f) ? v : slope * v;
        out[(size_t)(mbase + r) * NB_CLASSES + wave * 16 + ncol] = v;
    }
}

// ---------------------------------------------------------------------------
// Launch: inputs in setup_inputs() order:
//   d_in[0] = seq        (fp32, 10000*100*256)
//   d_in[1] = graph_len  (int32, constant 100 -- shapes are static, unused)
//   d_in[2] = W          (fp32, 128*256)
//   d_in[3] = b          (fp32, 128)
//   d_in[4] = prelu_a    (fp32, 1)
//   d_out   = [10000, 128] fp32
//   d_ws    = pooled scratch, 10000*256*4 = 10.24 MB
// ---------------------------------------------------------------------------
extern "C" void kernel_launch(void* const* d_in, const int* in_sizes, int n_in,
                              void* d_out, int out_size, void* d_ws, size_t ws_size,
                              hipStream_t stream) {
    const float* seq     = (const float*)d_in[0];
    const float* W       = (const float*)d_in[2];
    const float* bias    = (const float*)d_in[3];
    const float* prelu_a = (const float*)d_in[4];
    float*       outp    = (float*)d_out;
    float*       pooled  = (float*)d_ws;

    // 4 graphs per block -> 2500 blocks of 256 threads.
    pool_sum_kernel<<<NUM_GRAPHS / 4, 256, 0, stream>>>(seq, pooled);

    // 625 M-tiles, 8 N-tile waves per block.
    fc_prelu_wmma_kernel<<<NUM_GRAPHS / 16, 256, 0, stream>>>(
        pooled, W, bias, prelu_a, outp);
}